// GINModel_29411936043504
// MI455X (gfx1250) — compile-verified
//
#include <hip/hip_runtime.h>

#define N_NODES   50000
#define N_EDGES   800000
#define HIDDEN    64
#define DEPTH     3
#define NUM_GRAPHS 256
#define BN_EPS    1e-5f

#define N_TILES   (N_NODES / 16)          // 3125 M-tiles of 16 rows
#define WAVE_TASKS ((N_TILES + 1) / 2)    // 1563 (each wave does 2 tiles)

typedef float v2f __attribute__((ext_vector_type(2)));
typedef float v8f __attribute__((ext_vector_type(8)));

// ---------------- utility kernels ----------------

__global__ __launch_bounds__(256) void zero_kernel(float* __restrict__ p, int n) {
    int i = blockIdx.x * blockDim.x + threadIdx.x;
    if (i < n) p[i] = 0.0f;
}

__global__ __launch_bounds__(256) void copy_kernel(float* __restrict__ dst,
                                                   const float* __restrict__ src, int n) {
    int i = blockIdx.x * blockDim.x + threadIdx.x;
    if (i < n) dst[i] = src[i];
}

// h[dst[e]][:] += x[src[e]][:]   (h pre-initialized to x -> h = x + agg)
// One wave per edge: float2 gather per lane, two f32 atomic adds.
__global__ __launch_bounds__(256) void scatter_kernel(float* __restrict__ h,
                                                      const float* __restrict__ x,
                                                      const int* __restrict__ esrc,
                                                      const int* __restrict__ edst) {
    int t = blockIdx.x * blockDim.x + threadIdx.x;
    int e = t >> 5;
    if (e >= N_EDGES) return;
    int lane = t & 31;
    int s = esrc[e];
    int d = edst[e];
    v2f xv = *(const v2f*)(x + s * HIDDEN + lane * 2);
    float* hp = h + d * HIDDEN + lane * 2;
    atomicAdd(hp,     xv.x);
    atomicAdd(hp + 1, xv.y);
}

// ---------------- WMMA GEMM: out = relu(A[50000,64] @ W[64,64] + bias) ----------------
// fp32 WMMA (V_WMMA_F32_16X16X4_F32): exact fp32 math; workload is memory-bound,
// so no reason to quantize. Each wave computes TWO 16-row M tiles x all 64 columns:
// every B fragment (single ds_load_b64 from pair-packed LDS) feeds two WMMAs.

__global__ __launch_bounds__(256) void gemm_bias_relu(float* __restrict__ out,
                                                      const float* __restrict__ A,
                                                      const float* __restrict__ W,
                                                      const float* __restrict__ bias) {
    // Pair-packed weights: lw2[p*64 + n] = { W[2p][n], W[2p+1][n] }  (16 KB)
    __shared__ v2f lw2[32 * HIDDEN];
    const int tid = threadIdx.x;
    #pragma unroll
    for (int i = 0; i < 8; ++i) {
        int idx = i * 256 + tid;          // 0..2047
        int p = idx >> 6;
        int n = idx & 63;
        v2f w;
        w.x = W[(2 * p) * HIDDEN + n];
        w.y = W[(2 * p + 1) * HIDDEN + n];
        lw2[idx] = w;
    }
    __syncthreads();

    const int wave = tid >> 5;
    const int lane = tid & 31;
    const int task = blockIdx.x * 8 + wave;          // wave-uniform
    if (task >= WAVE_TASKS) return;                  // uniform: EXEC stays all-ones

    const int mtA = task;
    int mtB = task + WAVE_TASKS;
    if (mtB >= N_TILES) mtB = mtA;                   // tail clamp: duplicate work, benign

    const int col = lane & 15;                       // N within tile / A row within tile
    const int kh  = lane >> 4;                       // lane half -> K sub-pair
    const float* arow0 = A + (mtA * 16 + col) * HIDDEN + kh * 2;
    const float* arow1 = A + (mtB * 16 + col) * HIDDEN + kh * 2;

    v8f acc[8] = {v8f{}, v8f{}, v8f{}, v8f{}, v8f{}, v8f{}, v8f{}, v8f{}};

    #pragma unroll
    for (int k0 = 0; k0 < HIDDEN; k0 += 4) {
        // A frag: lane half h, component r -> K = k0 + 2h + r  (ISA 16x4 f32 layout)
        v2f a0 = *(const v2f*)(arow0 + k0);
        v2f a1 = *(const v2f*)(arow1 + k0);
        const v2f* wp = &lw2[(k0 / 2 + kh) * HIDDEN + col];
        #pragma unroll
        for (int nt = 0; nt < 4; ++nt) {
            v2f b = wp[nt * 16];                     // {W[k0+2h][n], W[k0+2h+1][n]}
            acc[nt] = __builtin_amdgcn_wmma_f32_16x16x4_f32(
                false, a0, false, b, (short)0, acc[nt], false, false);
            acc[4 + nt] = __builtin_amdgcn_wmma_f32_16x16x4_f32(
                false, a1, false, b, (short)0, acc[4 + nt], false, false);
        }
    }

    // C/D layout: VGPR r -> row mt*16 + kh*8 + r, column nt*16 + col
    float bv[4];
    #pragma unroll
    for (int nt = 0; nt < 4; ++nt) bv[nt] = bias[nt * 16 + col];

    const int rbaseA = mtA * 16 + kh * 8;
    const int rbaseB = mtB * 16 + kh * 8;
    #pragma unroll
    for (int r = 0; r < 8; ++r) {
        float* orowA = out + (rbaseA + r) * HIDDEN;
        float* orowB = out + (rbaseB + r) * HIDDEN;
        #pragma unroll
        for (int nt = 0; nt < 4; ++nt) {
            float vA = acc[nt][r] + bv[nt];
            float vB = acc[4 + nt][r] + bv[nt];
            orowA[nt * 16 + col] = vA > 0.0f ? vA : 0.0f;
            orowB[nt * 16 + col] = vB > 0.0f ? vB : 0.0f;
        }
    }
}

// ---------------- BatchNorm statistics: per-column sum & sum of squares ----------------
// Grid stride is a multiple of 64, so thread tid always touches column tid%64.

__global__ __launch_bounds__(256) void bn_stats(const float* __restrict__ h,
                                                float* __restrict__ csum,
                                                float* __restrict__ csum2) {
    __shared__ float ls[256];
    __shared__ float ls2[256];
    const int tid = threadIdx.x;
    float s = 0.0f, s2 = 0.0f;
    const int total = N_NODES * HIDDEN;
    const int stride = gridDim.x * blockDim.x;       // multiple of 64
    for (int i = blockIdx.x * blockDim.x + tid; i < total; i += stride) {
        float v = h[i];
        s += v;
        s2 += v * v;
    }
    ls[tid] = s;
    ls2[tid] = s2;
    __syncthreads();
    if (tid < 64) {
        s  = ls[tid]  + ls[tid + 64]  + ls[tid + 128]  + ls[tid + 192];
        s2 = ls2[tid] + ls2[tid + 64] + ls2[tid + 128] + ls2[tid + 192];
        atomicAdd(&csum[tid], s);
        atomicAdd(&csum2[tid], s2);
    }
}

__global__ __launch_bounds__(256) void bn_relu(float* __restrict__ out,
                                               const float* __restrict__ h,
                                               const float* __restrict__ csum,
                                               const float* __restrict__ csum2,
                                               const float* __restrict__ gamma,
                                               const float* __restrict__ beta) {
    int i = blockIdx.x * blockDim.x + threadIdx.x;
    if (i >= N_NODES * HIDDEN) return;
    int c = i & 63;
    const float invN = 1.0f / (float)N_NODES;
    float mean = csum[c] * invN;
    float var  = csum2[c] * invN - mean * mean;
    float v = (h[i] - mean) * rsqrtf(var + BN_EPS) * gamma[c] + beta[c];
    out[i] = v > 0.0f ? v : 0.0f;
}

// ---------------- segment-sum pooling + final projection ----------------

__global__ __launch_bounds__(256) void pool_kernel(float* __restrict__ pooled,
                                                   const float* __restrict__ x,
                                                   const int* __restrict__ batch) {
    int i = blockIdx.x * blockDim.x + threadIdx.x;
    if (i >= N_NODES * HIDDEN) return;
    int node = i >> 6;
    int c = i & 63;
    atomicAdd(&pooled[batch[node] * HIDDEN + c], x[i]);
}

// out[g] = pooled[g,:] . Wf + bf     (one wave per graph, wave32 shuffle reduce)
__global__ __launch_bounds__(256) void final_kernel(float* __restrict__ out,
                                                    const float* __restrict__ pooled,
                                                    const float* __restrict__ Wf,
                                                    const float* __restrict__ bf) {
    const int lane = threadIdx.x & 31;
    const int g = blockIdx.x * 8 + (threadIdx.x >> 5);
    const float* pg = pooled + g * HIDDEN;
    float p = pg[lane] * Wf[lane] + pg[lane + 32] * Wf[lane + 32];
    #pragma unroll
    for (int o = 16; o > 0; o >>= 1) p += __shfl_xor(p, o, 32);
    if (lane == 0) out[g] = p + bf[0];
}

// ---------------- host-side orchestration ----------------

extern "C" void kernel_launch(void* const* d_in, const int* in_sizes, int n_in,
                              void* d_out, int out_size, void* d_ws, size_t ws_size,
                              hipStream_t stream) {
    (void)in_sizes; (void)n_in; (void)out_size; (void)ws_size;

    const float* x     = (const float*)d_in[0];
    const int*   edge  = (const int*)d_in[1];
    const int*   batch = (const int*)d_in[2];
    const float* W1    = (const float*)d_in[3];
    const float* b1    = (const float*)d_in[4];
    const float* W2    = (const float*)d_in[5];
    const float* b2    = (const float*)d_in[6];
    const float* gamma = (const float*)d_in[7];
    const float* beta  = (const float*)d_in[8];
    const float* Wf    = (const float*)d_in[9];
    const float* bf    = (const float*)d_in[10];
    float* out = (float*)d_out;

    const size_t NH = (size_t)N_NODES * HIDDEN;           // 3.2M floats
    float* bufA   = (float*)d_ws;
    float* bufB   = bufA + NH;
    float* csum   = bufB + NH;
    float* csum2  = csum + HIDDEN;
    float* pooled = csum2 + HIDDEN;

    const int* esrc = edge;
    const int* edst = edge + N_EDGES;

    const int nh_n       = (int)NH;
    const int nh_blocks  = (nh_n + 255) / 256;
    const int sc_blocks  = (N_EDGES * 32 + 255) / 256;    // one wave per edge
    const int gemm_blocks = (WAVE_TASKS + 7) / 8;         // 8 waves per block

    const float* xin = x;
    for (int i = 0; i < DEPTH; ++i) {
        copy_kernel<<<nh_blocks, 256, 0, stream>>>(bufA, xin, nh_n);
        scatter_kernel<<<sc_blocks, 256, 0, stream>>>(bufA, xin, esrc, edst);
        gemm_bias_relu<<<gemm_blocks, 256, 0, stream>>>(bufB, bufA,
                                                        W1 + i * HIDDEN * HIDDEN,
                                                        b1 + i * HIDDEN);
        gemm_bias_relu<<<gemm_blocks, 256, 0, stream>>>(bufA, bufB,
                                                        W2 + i * HIDDEN * HIDDEN,
                                                        b2 + i * HIDDEN);
        zero_kernel<<<1, 128, 0, stream>>>(csum, 2 * HIDDEN);
        bn_stats<<<512, 256, 0, stream>>>(bufA, csum, csum2);
        bn_relu<<<nh_blocks, 256, 0, stream>>>(bufB, bufA, csum, csum2,
                                               gamma + i * HIDDEN, beta + i * HIDDEN);
        xin = bufB;
    }

    zero_kernel<<<(NUM_GRAPHS * HIDDEN + 255) / 256, 256, 0, stream>>>(pooled,
                                                                       NUM_GRAPHS * HIDDEN);
    pool_kernel<<<nh_blocks, 256, 0, stream>>>(pooled, xin, batch);
    final_kernel<<<NUM_GRAPHS / 8, 256, 0, stream>>>(out, pooled, Wf, bf);
}